// RandomizedHadamardTransform_26096221291226
// MI455X (gfx1250) — compile-verified
//
#include <hip/hip_runtime.h>
#include <stdint.h>

// Randomized Hadamard Transform for MI455X (gfx1250).
// H4096 = H16 (x) H16 (x) H16  -> three stages of 16x16 f32 WMMA tile matmuls
// (v_wmma_f32_16x16x4_f32, K=16 via 4 chained WMMAs) done in-place in LDS.
// Memory-bound: ~320 MB HBM traffic -> ~14 us floor at 23.3 TB/s.

typedef float v2f __attribute__((ext_vector_type(2)));
typedef float v4f __attribute__((ext_vector_type(4)));
typedef float v8f __attribute__((ext_vector_type(8)));

#define PAD_DIM 4096
#define OUT_DIM 1024
#define WAVES   4
#define THREADS (WAVES * 32)

// XOR swizzle: kills stride-16 (column-major) 4-way bank conflicts while
// preserving float4 contiguity/alignment (only flips word-index bits [3:2]).
__device__ __forceinline__ int swz(int idx) {
    return idx ^ (((idx >> 6) & 3) << 2);
}

// H16[i][j] = (-1)^popcount(i & j)
__device__ __forceinline__ float hval(int i, int j) {
    return (__popc(i & j) & 1) ? -1.0f : 1.0f;
}

// One FWHT stage over one 16-axis of the row, as 16 in-place 16x16 WMMA tiles.
// Tile element (c, n) lives at LDS word  t*tileStride + c*cStride + n*nStride.
// D = H16(A) x B(tile); H16 is symmetric so no transpose bookkeeping needed.
__device__ __forceinline__ void fwht_stage(float* buf, const v2f hA[4],
                                           int half, int nc,
                                           int tileStride, int cStride, int nStride)
{
#pragma unroll 1
    for (int t = 0; t < 16; ++t) {
        const int base = t * tileStride + nc * nStride;

        // B operand: 16x16 tile, K rows per ISA layout:
        //   VGPR0 -> K = {0 (lanes 0-15), 2 (lanes 16-31)}, VGPR1 -> K = {1, 3}
        v2f b[4];
#pragma unroll
        for (int k = 0; k < 4; ++k) {
            const int K0 = 4 * k + 2 * half;
            b[k].x = buf[swz(base + K0 * cStride)];
            b[k].y = buf[swz(base + (K0 + 1) * cStride)];
        }

        v8f acc = {0.f, 0.f, 0.f, 0.f, 0.f, 0.f, 0.f, 0.f};
#pragma unroll
        for (int k = 0; k < 4; ++k) {
            // (neg_a, A, neg_b, B, c_mod, C, reuse_a, reuse_b)
            acc = __builtin_amdgcn_wmma_f32_16x16x4_f32(
                    false, hA[k], false, b[k], (short)0, acc, false, false);
        }

        // D layout: VGPR v -> M = v (lanes 0-15) / v+8 (lanes 16-31), N = lane&15.
        // In-place write-back: all lanes' ds_loads precede these ds_stores in
        // program order and LDS ops from one wave are processed in order.
#pragma unroll
        for (int v = 0; v < 8; ++v) {
            const int M = v + 8 * half;
            buf[swz(base + M * cStride)] = acc[v];
        }
    }
}

__global__ __launch_bounds__(THREADS)
void rht_fwht_wmma_kernel(const float* __restrict__ x,
                          const float* __restrict__ signs,
                          const long long* __restrict__ sel,
                          float* __restrict__ out,
                          int nrows)
{
    __shared__ float lds[WAVES * PAD_DIM];   // 64 KB: one 16 KB row per wave

    const int lane = threadIdx.x & 31;
    const int wave = threadIdx.x >> 5;
    int row = blockIdx.x * WAVES + wave;
    if (row >= nrows) row = nrows - 1;       // wave-uniform clamp (benign dup)

    float* buf = &lds[wave * PAD_DIM];

    const int half = lane >> 4;              // 0: lanes 0-15, 1: lanes 16-31
    const int nc   = lane & 15;

    // Preload H16 into the WMMA A-operand layout (constant for all stages):
    // lane M = lane&15; chunk k covers columns 4k..4k+3 with
    //   VGPR0 -> col {4k, 4k+2}, VGPR1 -> col {4k+1, 4k+3} (by half-wave).
    v2f hA[4];
#pragma unroll
    for (int k = 0; k < 4; ++k) {
        const int col = 4 * k + 2 * half;
        hA[k].x = hval(nc, col);
        hA[k].y = hval(nc, col + 1);
    }

    // ---- HBM -> LDS: coalesced float4 loads, fold in signs ----
    const float* xrow = x + (size_t)row * PAD_DIM;
#pragma unroll
    for (int it = 0; it < PAD_DIM / (32 * 4); ++it) {     // 32 iters
        const int i4 = (it * 32 + lane) * 4;
        v4f v = *(const v4f*)(xrow + i4);
        const v4f s = *(const v4f*)(signs + i4);          // L2/WGP$-resident
        v *= s;
        *(v4f*)(buf + swz(i4)) = v;                       // ds_store_b128
    }
    __syncthreads();

    // index n = a*256 + b*16 + c
    // Stage 0: contract c -> tile base a*256, c stride 1,   col(b) stride 16
    fwht_stage(buf, hA, half, nc, 256, 1, 16);
    __syncthreads();
    // Stage 1: contract b -> tile base a*256, b stride 16,  col(i) stride 1
    fwht_stage(buf, hA, half, nc, 256, 16, 1);
    __syncthreads();
    // Stage 2: contract a -> tile base q*16,  a stride 256, col stride 1
    fwht_stage(buf, hA, half, nc, 16, 256, 1);
    __syncthreads();

    // ---- gather selected columns, scale by 1/sqrt(4096), coalesced stores ----
    const float scale = 0.015625f;           // 1/64
    float* orow = out + (size_t)row * OUT_DIM;
#pragma unroll
    for (int it = 0; it < OUT_DIM / 32; ++it) {
        const int k = it * 32 + lane;
        const int idx = (int)sel[k];         // int64 indices, values < 4096
        orow[k] = buf[swz(idx)] * scale;
    }
}

extern "C" void kernel_launch(void* const* d_in, const int* in_sizes, int n_in,
                              void* d_out, int out_size, void* d_ws, size_t ws_size,
                              hipStream_t stream) {
    const float*     x     = (const float*)d_in[0];
    const float*     signs = (const float*)d_in[1];
    const long long* sel   = (const long long*)d_in[2];
    float*           out   = (float*)d_out;

    const int nrows = in_sizes[0] / PAD_DIM;              // 16384
    const int blocks = (nrows + WAVES - 1) / WAVES;       // 4096
    rht_fwht_wmma_kernel<<<blocks, THREADS, 0, stream>>>(x, signs, sel, out, nrows);
}